// Block_3212635537783
// MI455X (gfx1250) — compile-verified
//
#include <hip/hip_runtime.h>
#include <math.h>

// ---------------------------------------------------------------------------
// CDNA5 (gfx1250) transformer block, fp32 end-to-end using V_WMMA_F32_16X16X4_F32
// B=4 T=2048 C=768 H=12 D=64  (~160 GFLOP, compute bound -> WMMA everywhere)
// Round 2: pre-transposed weights so every B-fragment K-pair is a single
// global_load_b64 (halves VMEM issue in all GEMM inner loops).
// ---------------------------------------------------------------------------

typedef float v2f __attribute__((ext_vector_type(2)));
typedef float v8f __attribute__((ext_vector_type(8)));

#define BN   4
#define TT   2048
#define CC   768
#define HH   12
#define DD   64
#define FF   3072
#define NTOK 8192
#define LDA  772      // LDS row stride (768+4) -> rotates banks, no 16-way conflicts
#define SROW 2052     // score-row stride (2048+4)

// D = A(16x4) * B(4x16) + C, fp32.
// (neg_a, A, neg_b, B, c_mod, C, reuse_a, reuse_b)
__device__ __forceinline__ v8f wmma4(v2f a, v2f b, v8f c) {
  return __builtin_amdgcn_wmma_f32_16x16x4_f32(false, a, false, b, (short)0, c,
                                               false, false);
}

// ===========================================================================
// Kernel 0: LDS-tiled batched transpose  src:(K x N) -> dst:(N x K)
// ===========================================================================
__global__ __launch_bounds__(256) void transpose_kernel(
    const float* __restrict__ src, float* __restrict__ dst, int K, int N) {
  __shared__ float tile[32][33];
  const size_t bofs = (size_t)blockIdx.z * K * N;
  src += bofs;
  dst += bofs;
  const int k0 = blockIdx.y * 32, n0 = blockIdx.x * 32;
  const int tx = threadIdx.x & 31, ty = threadIdx.x >> 5;  // 32x8
  for (int i = ty; i < 32; i += 8)
    tile[i][tx] = src[(size_t)(k0 + i) * N + n0 + tx];
  __syncthreads();
  for (int i = ty; i < 32; i += 8)
    dst[(size_t)(n0 + i) * K + k0 + tx] = tile[tx][i];
}

// ===========================================================================
// Kernel 1: per-head QKV projection.
//   x:(8192,768) @ Wt:[o][h][d][c] -> q,k:(B,H,T,D)  v:(B,H,D,T) (transposed!)
// ===========================================================================
__global__ __launch_bounds__(256) void qkv_kernel(
    const float* __restrict__ x, const float* __restrict__ wqkvt,
    float* __restrict__ qo, float* __restrict__ ko, float* __restrict__ vo) {
  __shared__ float xs[64 * LDA];
  const int m0 = blockIdx.x * 64;
  for (int i = threadIdx.x; i < 64 * (CC / 4); i += 256) {
    const int r = i / (CC / 4), c4 = i % (CC / 4);
    const float4 src = ((const float4*)(x + (m0 + r) * CC))[c4];
    *(float4*)(&xs[r * LDA + c4 * 4]) = src;
  }
  __syncthreads();
  const int wv = threadIdx.x >> 5, l = threadIdx.x & 31;
  const int n16 = l & 15, kq = (l >> 4) << 1;
  // 144 strip-tasks: 4 row-blocks x (3 outputs x 12 heads), each 16 rows x 64 cols
  for (int task = wv; task < 144; task += 8) {
    const int rb = task / 36, oh = task % 36;
    const int o = oh / 12, head = oh % 12;
    const float* Wt = wqkvt + (size_t)oh * (DD * CC);  // row d, 768 contiguous k
    v8f a0 = {}, a1 = {}, a2 = {}, a3 = {};
    const float* arow = &xs[(rb * 16 + n16) * LDA];
    const float* brow = Wt + (size_t)n16 * CC;
    for (int k0 = 0; k0 < CC; k0 += 4) {
      const v2f a = *(const v2f*)(arow + k0 + kq);
      const v2f b0 = *(const v2f*)(brow + k0 + kq);
      const v2f b1 = *(const v2f*)(brow + 16 * CC + k0 + kq);
      const v2f b2 = *(const v2f*)(brow + 32 * CC + k0 + kq);
      const v2f b3 = *(const v2f*)(brow + 48 * CC + k0 + kq);
      a0 = wmma4(a, b0, a0); a1 = wmma4(a, b1, a1);
      a2 = wmma4(a, b2, a2); a3 = wmma4(a, b3, a3);
    }
    const int mbase = rb * 16 + ((l >> 4) << 3);
    if (o < 2) {
      float* OUT = (o == 0) ? qo : ko;
#pragma unroll
      for (int j = 0; j < 8; ++j) {
        const int tok = m0 + mbase + j;
        const int b = tok >> 11, t = tok & 2047;
        float* dst = OUT + (((size_t)(b * HH + head) * TT + t) * DD);
        dst[0 + n16] = a0[j];  dst[16 + n16] = a1[j];
        dst[32 + n16] = a2[j]; dst[48 + n16] = a3[j];
      }
    } else {  // v stored transposed: vo[(bh*DD + d)*TT + t]
#pragma unroll
      for (int j = 0; j < 8; ++j) {
        const int tok = m0 + mbase + j;
        const int b = tok >> 11, t = tok & 2047;
        float* base = vo + (size_t)(b * HH + head) * DD * TT + t;
        base[(size_t)(0 + n16) * TT] = a0[j];
        base[(size_t)(16 + n16) * TT] = a1[j];
        base[(size_t)(32 + n16) * TT] = a2[j];
        base[(size_t)(48 + n16) * TT] = a3[j];
      }
    }
  }
}

// ===========================================================================
// Kernel 2: causal attention (source quirks):
//   logits[t][s] = k[t].q[s] / sqrt(C), mask s<=t, softmax over s, O = P @ v.
// One workgroup per (b,h, 16-row t-tile); full 16x2048 score rows in LDS.
// v arrives transposed (D,T) so P@V B-fragments are contiguous b64 loads.
// ===========================================================================
__global__ __launch_bounds__(256) void attn_kernel(
    const float* __restrict__ q, const float* __restrict__ k,
    const float* __restrict__ vt, float* __restrict__ aout) {
  __shared__ float S[16 * SROW];
  __shared__ float red[256];
  __shared__ float rowmax[16], rowrcp[16];
  __shared__ float opart[16 * 64];
  const int ttile = blockIdx.x;            // 0..127
  const int bh = blockIdx.y;               // 0..47
  const int t0 = ttile * 16;
  const float* qp = q + (size_t)bh * TT * DD;
  const float* kp = k + (size_t)bh * TT * DD;
  const float* vp = vt + (size_t)bh * TT * DD;  // [d][t]
  const int wv = threadIdx.x >> 5, l = threadIdx.x & 31;
  const int n16 = l & 15, hi = l >> 4, kq = hi << 1;
  const int NT = ttile + 1, L = NT * 16;   // only s-tiles <= t needed
  const float scale = 0.036084391824352f;  // 1/sqrt(768)

  // ---- Phase A: S = K_tile(16x64) @ Q^T, scaled + causal-masked, into LDS ----
  v2f af[16];                              // k-rows fragment, reused for all s
#pragma unroll
  for (int ks = 0; ks < 16; ++ks)
    af[ks] = *(const v2f*)(kp + (t0 + n16) * DD + ks * 4 + kq);
  for (int st = wv; st < NT; st += 8) {
    const int s0 = st * 16;
    v8f acc = {};
#pragma unroll
    for (int ks = 0; ks < 16; ++ks) {
      const v2f b = *(const v2f*)(qp + (s0 + n16) * DD + ks * 4 + kq);
      acc = wmma4(af[ks], b, acc);
    }
    const int sg = s0 + n16;
#pragma unroll
    for (int j = 0; j < 8; ++j) {
      const int row = (hi << 3) + j;
      float vl = acc[j] * scale;
      if (sg > t0 + row) vl = -1e30f;      // causal: keep s <= t
      S[row * SROW + sg] = vl;
    }
  }
  __syncthreads();

  // ---- Phase B: two-pass softmax (unnormalized exp in LDS, 1/sum kept) ----
  {
    const int row = threadIdx.x >> 4, sub = threadIdx.x & 15;
    float m = -1e30f;
    for (int c = sub; c < L; c += 16) m = fmaxf(m, S[row * SROW + c]);
    red[threadIdx.x] = m;
    __syncthreads();
    if (sub == 0) {
      float mm = red[threadIdx.x];
      for (int i = 1; i < 16; ++i) mm = fmaxf(mm, red[threadIdx.x + i]);
      rowmax[row] = mm;
    }
    __syncthreads();
    const float mr = rowmax[row];
    float s = 0.f;
    for (int c = sub; c < L; c += 16) {
      const float e = __expf(S[row * SROW + c] - mr);
      S[row * SROW + c] = e;
      s += e;
    }
    red[threadIdx.x] = s;
    __syncthreads();
    if (sub == 0) {
      float ss = 0.f;
      for (int i = 0; i < 16; ++i) ss += red[threadIdx.x + i];
      rowrcp[row] = 1.0f / ss;
    }
  }
  __syncthreads();

  // ---- Phase C: O = P(16xL) @ V(Lx64); K-dim split across wave pairs ----
  const int tile = wv & 3, half = wv >> 2;
  const int ksteps = L >> 2, kh = ksteps >> 1;
  const int kbeg = half * kh, kend = kbeg + kh;
  v8f acc = {};
  const float* vrow = vp + (size_t)(tile * 16 + n16) * TT;  // contiguous in s
  for (int ks = kbeg; ks < kend; ++ks) {
    const v2f a = *(const v2f*)(&S[n16 * SROW + ks * 4 + kq]);
    const v2f b = *(const v2f*)(vrow + ks * 4 + kq);
    acc = wmma4(a, b, acc);
  }
  if (half == 1) {
#pragma unroll
    for (int j = 0; j < 8; ++j)
      opart[((hi << 3) + j) * 64 + tile * 16 + n16] = acc[j];
  }
  __syncthreads();
  if (half == 0) {
    const int bb = bh / HH, head = bh % HH;
#pragma unroll
    for (int j = 0; j < 8; ++j) {
      const int row = (hi << 3) + j;
      const float val =
          (acc[j] + opart[row * 64 + tile * 16 + n16]) * rowrcp[row];
      const int t = t0 + row;
      aout[((size_t)(bb * TT + t)) * CC + head * DD + tile * 16 + n16] = val;
    }
  }
}

// ===========================================================================
// Kernel 3/6: out = LayerNorm(xa + xb) * g + b   (one block per token)
// ===========================================================================
__global__ __launch_bounds__(256) void add_ln_kernel(
    const float* __restrict__ xa, const float* __restrict__ xb,
    const float* __restrict__ g, const float* __restrict__ bt,
    float* __restrict__ out) {
  __shared__ float red[256];
  const int tok = blockIdx.x;
  const float* pa = xa + (size_t)tok * CC;
  const float* pb = xb + (size_t)tok * CC;
  float vals[3];
  float s1 = 0.f;
#pragma unroll
  for (int i = 0; i < 3; ++i) {
    const int c = threadIdx.x + i * 256;
    vals[i] = pa[c] + pb[c];
    s1 += vals[i];
  }
  red[threadIdx.x] = s1;
  __syncthreads();
  for (int off = 128; off > 0; off >>= 1) {
    if (threadIdx.x < off) red[threadIdx.x] += red[threadIdx.x + off];
    __syncthreads();
  }
  const float mean = red[0] * (1.0f / CC);
  __syncthreads();
  float s2 = 0.f;
#pragma unroll
  for (int i = 0; i < 3; ++i) {
    const float d = vals[i] - mean;
    s2 += d * d;
  }
  red[threadIdx.x] = s2;
  __syncthreads();
  for (int off = 128; off > 0; off >>= 1) {
    if (threadIdx.x < off) red[threadIdx.x] += red[threadIdx.x + off];
    __syncthreads();
  }
  const float rstd = rsqrtf(red[0] * (1.0f / CC) + 1e-5f);
#pragma unroll
  for (int i = 0; i < 3; ++i) {
    const int c = threadIdx.x + i * 256;
    out[(size_t)tok * CC + c] = (vals[i] - mean) * rstd * g[c] + bt[c];
  }
}

// ===========================================================================
// Kernel 4: h = gelu_exact(x1 @ W1 + b1)   M=8192 N=3072 K=768, W1t:(N,K)
// ===========================================================================
__global__ __launch_bounds__(256) void mlp1_kernel(
    const float* __restrict__ xin, const float* __restrict__ W1t,
    const float* __restrict__ bias, float* __restrict__ hout) {
  __shared__ float xs[64 * LDA];
  const int m0 = blockIdx.x * 64;
  for (int i = threadIdx.x; i < 64 * (CC / 4); i += 256) {
    const int r = i / (CC / 4), c4 = i % (CC / 4);
    const float4 src = ((const float4*)(xin + (m0 + r) * CC))[c4];
    *(float4*)(&xs[r * LDA + c4 * 4]) = src;
  }
  __syncthreads();
  const int wv = threadIdx.x >> 5, l = threadIdx.x & 31;
  const int n16 = l & 15, kq = (l >> 4) << 1;
  for (int task = wv; task < 192; task += 8) {  // 4 rowblocks x 48 strips(64)
    const int rb = task / 48, nb = (task % 48) * 64;
    v8f a0 = {}, a1 = {}, a2 = {}, a3 = {};
    const float* arow = &xs[(rb * 16 + n16) * LDA];
    const float* brow = W1t + (size_t)(nb + n16) * CC;
    for (int k0 = 0; k0 < CC; k0 += 4) {
      const v2f a = *(const v2f*)(arow + k0 + kq);
      const v2f b0 = *(const v2f*)(brow + k0 + kq);
      const v2f b1 = *(const v2f*)(brow + (size_t)16 * CC + k0 + kq);
      const v2f b2 = *(const v2f*)(brow + (size_t)32 * CC + k0 + kq);
      const v2f b3 = *(const v2f*)(brow + (size_t)48 * CC + k0 + kq);
      a0 = wmma4(a, b0, a0); a1 = wmma4(a, b1, a1);
      a2 = wmma4(a, b2, a2); a3 = wmma4(a, b3, a3);
    }
    const int mbase = rb * 16 + ((l >> 4) << 3);
#pragma unroll
    for (int j = 0; j < 8; ++j) {
      float* dst = hout + (size_t)(m0 + mbase + j) * FF + nb + n16;
      float z;
      z = a0[j] + bias[nb + n16];
      dst[0] = 0.5f * z * (1.f + erff(z * 0.70710678118f));
      z = a1[j] + bias[nb + 16 + n16];
      dst[16] = 0.5f * z * (1.f + erff(z * 0.70710678118f));
      z = a2[j] + bias[nb + 32 + n16];
      dst[32] = 0.5f * z * (1.f + erff(z * 0.70710678118f));
      z = a3[j] + bias[nb + 48 + n16];
      dst[48] = 0.5f * z * (1.f + erff(z * 0.70710678118f));
    }
  }
}

// ===========================================================================
// Kernel 5: t2 = h @ W2 + b2   M=8192 N=768 K=3072, W2t:(N,K); K split on waves
// ===========================================================================
__global__ __launch_bounds__(256) void mlp2_kernel(
    const float* __restrict__ hin, const float* __restrict__ W2t,
    const float* __restrict__ bias, float* __restrict__ t2) {
  __shared__ float opart[64 * 64];
  const int m0 = blockIdx.x * 64, n0 = blockIdx.y * 64;
  const int wv = threadIdx.x >> 5, l = threadIdx.x & 31;
  const int n16 = l & 15, kq = (l >> 4) << 1;
  const int rb = wv & 3, half = wv >> 2;
  const int kbeg = half * 384, kend = kbeg + 384;  // 768 k-steps total
  v8f a0 = {}, a1 = {}, a2 = {}, a3 = {};
  const float* arow = hin + (size_t)(m0 + rb * 16 + n16) * FF;
  const float* brow = W2t + (size_t)(n0 + n16) * FF;
  for (int ks = kbeg; ks < kend; ++ks) {
    const int kk = ks * 4 + kq;
    const v2f a = *(const v2f*)(arow + kk);
    const v2f b0 = *(const v2f*)(brow + kk);
    const v2f b1 = *(const v2f*)(brow + (size_t)16 * FF + kk);
    const v2f b2 = *(const v2f*)(brow + (size_t)32 * FF + kk);
    const v2f b3 = *(const v2f*)(brow + (size_t)48 * FF + kk);
    a0 = wmma4(a, b0, a0); a1 = wmma4(a, b1, a1);
    a2 = wmma4(a, b2, a2); a3 = wmma4(a, b3, a3);
  }
  const int mb = rb * 16 + ((l >> 4) << 3);
  if (half == 1) {
#pragma unroll
    for (int j = 0; j < 8; ++j) {
      opart[(mb + j) * 64 + n16] = a0[j];
      opart[(mb + j) * 64 + 16 + n16] = a1[j];
      opart[(mb + j) * 64 + 32 + n16] = a2[j];
      opart[(mb + j) * 64 + 48 + n16] = a3[j];
    }
  }
  __syncthreads();
  if (half == 0) {
#pragma unroll
    for (int j = 0; j < 8; ++j) {
      float* dst = t2 + (size_t)(m0 + mb + j) * CC + n0 + n16;
      dst[0]  = a0[j] + opart[(mb + j) * 64 + n16]      + bias[n0 + n16];
      dst[16] = a1[j] + opart[(mb + j) * 64 + 16 + n16] + bias[n0 + 16 + n16];
      dst[32] = a2[j] + opart[(mb + j) * 64 + 32 + n16] + bias[n0 + 32 + n16];
      dst[48] = a3[j] + opart[(mb + j) * 64 + 48 + n16] + bias[n0 + 48 + n16];
    }
  }
}

// ===========================================================================
extern "C" void kernel_launch(void* const* d_in, const int* in_sizes, int n_in,
                              void* d_out, int out_size, void* d_ws,
                              size_t ws_size, hipStream_t stream) {
  (void)in_sizes; (void)n_in; (void)out_size; (void)ws_size;
  const float* x    = (const float*)d_in[0];
  const float* Wq   = (const float*)d_in[1];
  const float* Wk   = (const float*)d_in[2];
  const float* Wv   = (const float*)d_in[3];
  const float* ln1g = (const float*)d_in[4];
  const float* ln1b = (const float*)d_in[5];
  const float* W1   = (const float*)d_in[6];
  const float* b1   = (const float*)d_in[7];
  const float* W2   = (const float*)d_in[8];
  const float* b2   = (const float*)d_in[9];
  const float* ln2g = (const float*)d_in[10];
  const float* ln2b = (const float*)d_in[11];
  float* ws  = (float*)d_ws;
  float* out = (float*)d_out;

  // workspace layout (floats); h reuses q/k/v/a region after they're consumed
  float* q     = ws;                 // 6,291,456
  float* k     = ws + 6291456;       // 6,291,456
  float* v     = ws + 12582912;      // 6,291,456  (B,H,D,T transposed)
  float* a     = ws + 18874368;      // 6,291,456
  float* hbuf  = ws;                 // 25,165,824 (reuse: k4 runs after k2/k3)
  float* x1    = ws + 25165824;      // 6,291,456
  float* t2    = ws + 31457280;      // 6,291,456
  float* wqkvt = ws + 37748736;      // 1,769,472  [o][h][d][c]
  float* w1t   = ws + 39518208;      // 2,359,296  (3072 x 768)
  float* w2t   = ws + 41877504;      // 2,359,296  (768 x 3072)  -> ~177 MB

  // ---- weight transposes (cheap: 26 MB once per call) ----
  transpose_kernel<<<dim3(DD / 32, CC / 32, HH), 256, 0, stream>>>(
      Wq, wqkvt + 0 * (size_t)HH * DD * CC, CC, DD);
  transpose_kernel<<<dim3(DD / 32, CC / 32, HH), 256, 0, stream>>>(
      Wk, wqkvt + 1 * (size_t)HH * DD * CC, CC, DD);
  transpose_kernel<<<dim3(DD / 32, CC / 32, HH), 256, 0, stream>>>(
      Wv, wqkvt + 2 * (size_t)HH * DD * CC, CC, DD);
  transpose_kernel<<<dim3(FF / 32, CC / 32, 1), 256, 0, stream>>>(
      W1, w1t, CC, FF);
  transpose_kernel<<<dim3(CC / 32, FF / 32, 1), 256, 0, stream>>>(
      W2, w2t, FF, CC);

  qkv_kernel<<<dim3(NTOK / 64), 256, 0, stream>>>(x, wqkvt, q, k, v);
  attn_kernel<<<dim3(TT / 16, BN * HH), 256, 0, stream>>>(q, k, v, a);
  add_ln_kernel<<<dim3(NTOK), 256, 0, stream>>>(x, a, ln1g, ln1b, x1);
  mlp1_kernel<<<dim3(NTOK / 64), 256, 0, stream>>>(x1, w1t, b1, hbuf);
  mlp2_kernel<<<dim3(NTOK / 64, CC / 64), 256, 0, stream>>>(hbuf, w2t, b2, t2);
  add_ln_kernel<<<dim3(NTOK), 256, 0, stream>>>(x1, t2, ln2g, ln2b, out);
}